// PaillerMLP_40621800685756
// MI455X (gfx1250) — compile-verified
//
#include <hip/hip_runtime.h>

// CDNA5 (gfx1250, wave32) WMMA fragment types
typedef __attribute__((ext_vector_type(2))) float v2f;   // 16x4 / 4x16 f32 fragment: 2 VGPRs
typedef __attribute__((ext_vector_type(8))) float v8f;   // 16x16 f32 accumulator: 8 VGPRs

// out[r] = act( sum_k W[r][k] * x[k] + b[r] ),  act = square or identity.
//
// Block = 256 threads = 8 waves. Each block owns a 16-row tile of W; the 8
// waves split the K dimension 8 ways, each running a chain of
// V_WMMA_F32_16X16X4_F32 ops:
//   A (16x4)  : x[k..k+3] in matrix row M=0 only  -> lanes 0,16 carry data
//   B (4x16)  : W-tile transposed; lane j holds W[rowBase+j][k+(lane>>4)*2 ..+1]
//   D (16x16) : column N=j accumulates row j's dot product; the 16 results
//               land in accumulator VGPR0, lanes 0..15.
// Cross-wave split-K partials are reduced through LDS, then bias+activation.
//
// The loop trip count is derived ONLY from K (an SGPR), so the k-loop is
// wave-uniform: no EXEC masking, plain s_cmp/s_cbranch, unrollable.
__global__ __launch_bounds__(256)
void matvec_wmma_f32(const float* __restrict__ W,
                     const float* __restrict__ x,
                     const float* __restrict__ b,
                     float* __restrict__ out,
                     int M, int K, int squareAct)
{
    const int tid  = threadIdx.x;
    const int lane = tid & 31;
    const int wave = tid >> 5;                 // 0..7
    const int rowBase = blockIdx.x << 4;       // 16 rows per block

    // Split-K: each wave reduces K/8 of the dot product.
    // groups of 8 WMMA steps: steps = (K/8)/4 /8 groups -> K/256 groups.
    // K = 4096 -> 16 groups, K = 8192 -> 32 groups (both exact).
    const int kChunk  = K >> 3;                // floats per wave
    const int nGroups = kChunk >> 5;           // 8 wmma steps (32 floats) per group
    const int kBegin  = wave * kChunk;

    // B-fragment (W) addressing: lane supplies row rowBase+(lane&15),
    // K-pair offset (lane>>4)*2. Clamp row for the ragged last tile (M=1000);
    // clamped lanes read valid memory and their results are simply not stored.
    int r = rowBase + (lane & 15);
    if (r >= M) r = M - 1;
    const float* wp = W + (size_t)r * (size_t)K + (size_t)(kBegin + ((lane >> 4) << 1));

    // A-fragment (x): only lanes 0 and 16 are nonzero (matrix row M=0).
    const bool   xlane = (lane & 15) == 0;     // lanes 0 and 16
    const float* xp    = x + kBegin + ((lane >> 4) << 1);

    v8f acc = {};
    for (int g = 0; g < nGroups; ++g) {
#pragma unroll
        for (int u = 0; u < 8; ++u) {
            // 8 bytes/lane of W per step = 256B/wave/instruction; the 8
            // unrolled steps use immediate offsets off one base pointer and
            // fully consume each 128B line -> HBM sees exactly one pass of W.
            v2f wfrag = *(const v2f*)(wp + (u << 2));
            v2f xv    = *(const v2f*)(xp + (u << 2));
            v2f afrag;                          // branch-free: EXEC stays all-1s
            afrag.x = xlane ? xv.x : 0.0f;
            afrag.y = xlane ? xv.y : 0.0f;
            acc = __builtin_amdgcn_wmma_f32_16x16x4_f32(
                /*neg_a=*/false, afrag,
                /*neg_b=*/false, wfrag,
                /*c_mod=*/(short)0, acc,
                /*reuse_a=*/false, /*reuse_b=*/false);
        }
        wp += 32;
        xp += 32;
    }

    // acc[0] (VGPR0) lanes 0..15 hold the 16 per-row partial sums of this wave.
    __shared__ float red[8][16];
    if (lane < 16) red[wave][lane] = acc[0];
    __syncthreads();

    if (tid < 16) {
        const int row = rowBase + tid;
        if (row < M) {
            float s = 0.0f;
#pragma unroll
            for (int wv = 0; wv < 8; ++wv) s += red[wv][tid];
            s += b[row];
            if (squareAct) s *= s;
            out[row] = s;
        }
    }
}

extern "C" void kernel_launch(void* const* d_in, const int* in_sizes, int n_in,
                              void* d_out, int out_size, void* d_ws, size_t ws_size,
                              hipStream_t stream)
{
    (void)in_sizes; (void)n_in; (void)out_size; (void)ws_size;

    constexpr int D_IN  = 4096;
    constexpr int D_HID = 8192;
    constexpr int D_OUT = 1000;

    const float* x  = (const float*)d_in[0];
    const float* W1 = (const float*)d_in[1];
    const float* b1 = (const float*)d_in[2];
    const float* W2 = (const float*)d_in[3];
    const float* b2 = (const float*)d_in[4];
    const float* W3 = (const float*)d_in[5];
    const float* b3 = (const float*)d_in[6];
    float* outp = (float*)d_out;

    // Scratch: h1 and h2 activations (8192 floats each = 64 KB total).
    float* h1 = (float*)d_ws;
    float* h2 = h1 + D_HID;

    // Layer 1: h1 = (W1 @ x + b1)^2          [8192 x 4096]
    matvec_wmma_f32<<<D_HID / 16, 256, 0, stream>>>(W1, x,  b1, h1, D_HID, D_IN,  1);
    // Layer 2: h2 = (W2 @ h1 + b2)^2         [8192 x 8192]
    matvec_wmma_f32<<<D_HID / 16, 256, 0, stream>>>(W2, h1, b2, h2, D_HID, D_HID, 1);
    // Layer 3: out = W3 @ h2 + b3            [1000 x 8192]
    matvec_wmma_f32<<<(D_OUT + 15) / 16, 256, 0, stream>>>(W3, h2, b3, outp, D_OUT, D_HID, 0);
}